// ScaledDotProductAttention_75505525064375
// MI455X (gfx1250) — compile-verified
//
#include <hip/hip_runtime.h>
#include <hip/hip_bf16.h>

// Problem constants from the reference
#define TT 4
#define BB 16
#define LL 1024
#define DD 256
#define NCHUNK 8            // L split into 8 chunks of 128 rows
#define LCHUNK 128
#define OUT_ELEMS (TT * BB * LL * DD)   // 16,777,216 per output tensor

typedef float v2f __attribute__((ext_vector_type(2)));
typedef float v8f __attribute__((ext_vector_type(8)));
typedef float f4  __attribute__((ext_vector_type(4)));

// ---------------------------------------------------------------------------
// Kernel 1: attn = mask ? 0 : k*v  (streamed out NT), and partial L-sums via
// chained V_WMMA_F32_16X16X4_F32 with an all-ones A matrix (exact f32 math:
// C[0,n] accumulates sum_k B[k,n]; A layout irrelevant since A==1 everywhere,
// B K-slot placement irrelevant since every element contributes once to its
// column sum — only lane&15 <-> d column mapping matters).
// One wave handles one (t,b, 128-row l-chunk, 16-col d-group) tile.
//   waves = 64 (t,b) * 8 chunks * 16 dgroups = 8192  -> 1024 blocks of 256 thr
// ---------------------------------------------------------------------------
__global__ __launch_bounds__(256) void snn_attn_partial_kernel(
    const float* __restrict__ kp, const float* __restrict__ vp,
    const unsigned char* __restrict__ maskp,       // jax bool -> 1 byte
    float* __restrict__ attn_out, float* __restrict__ partial)
{
    const int tid  = blockIdx.x * blockDim.x + threadIdx.x;
    const int wave = tid >> 5;
    const int lane = threadIdx.x & 31;
    const int col  = lane & 15;          // d within group (== WMMA column N)
    const int half = lane >> 4;          // which pair of K rows this lane feeds

    const int dgroup = wave & 15;        // 16 groups of 16 d
    const int chunk  = (wave >> 4) & 7;  // 8 chunks of 128 l
    const int tb     = wave >> 7;        // flattened (t,b) in [0,64)

    const int d0      = dgroup * 16 + col;
    const int lbase   = chunk * LCHUNK;
    const int rowbase = tb * LL;         // row index base for this (t,b)

    v8f c = {};                          // 16x16 f32 accumulator
    v2f a; a.x = 1.0f; a.y = 1.0f;       // all-ones A (16x4)

    for (int i = 0; i < LCHUNK / 4; ++i) {
        const int r0 = lbase + i * 4 + half * 2;     // this lane's two rows
        const int o0 = (rowbase + r0) * DD + d0;
        const int o1 = o0 + DD;

        // gfx1250 global_prefetch: pull next tile (32 rows ahead) toward L2
        // while the chained WMMA accumulation serializes this wave.
        if (i < (LCHUNK / 4) - 8) {
            __builtin_prefetch(kp + o0 + 32 * DD, 0, 1);
            __builtin_prefetch(vp + o0 + 32 * DD, 0, 1);
        }

        // k/v are consumed exactly once -> non-temporal loads
        const float k0 = __builtin_nontemporal_load(kp + o0);
        const float v0 = __builtin_nontemporal_load(vp + o0);
        const float k1 = __builtin_nontemporal_load(kp + o1);
        const float v1 = __builtin_nontemporal_load(vp + o1);
        const unsigned char m0 = maskp[rowbase + r0];
        const unsigned char m1 = maskp[rowbase + r0 + 1];

        v2f b;                            // B tile: 4 rows x 16 cols of attn
        b.x = m0 ? 0.0f : k0 * v0;
        b.y = m1 ? 0.0f : k1 * v1;

        // attn is a write-once 64MB output stream -> non-temporal stores
        __builtin_nontemporal_store(b.x, attn_out + o0);
        __builtin_nontemporal_store(b.y, attn_out + o1);

        // C = 1 * B + C : folds 4 l-rows into the 16 column sums
        c = __builtin_amdgcn_wmma_f32_16x16x4_f32(
                /*neg_a=*/false, a, /*neg_b=*/false, b,
                /*c_mod=*/(short)0, c, /*reuse_a=*/false, /*reuse_b=*/false);
    }

    // C row M=0 lives in VGPR0 of lanes 0..15 (N = lane): the column sums
    if (lane < 16) {
        partial[(tb * NCHUNK + chunk) * DD + dgroup * 16 + col] = c[0];
    }
}

// ---------------------------------------------------------------------------
// Kernel 2: combine the 8 chunk partials per (t,b,d), then run the LIF gate
// sequentially over T (T=4, tau=2, hard reset, threshold 1).
// B*D = 4096 independent lanes -> 16 blocks of 256.
// ---------------------------------------------------------------------------
__global__ __launch_bounds__(256) void snn_lif_kernel(
    const float* __restrict__ partial, float* __restrict__ spikes)
{
    const int tid = blockIdx.x * blockDim.x + threadIdx.x;
    if (tid >= BB * DD) return;
    const int b = tid >> 8;       // DD == 256
    const int d = tid & (DD - 1);

    float vs = 0.0f;
    for (int t = 0; t < TT; ++t) {
        const int tb = t * BB + b;
        float kv = 0.0f;
        #pragma unroll
        for (int cc = 0; cc < NCHUNK; ++cc)
            kv += partial[(tb * NCHUNK + cc) * DD + d];
        const float h = 0.5f * (vs + kv);          // v + (x - v)/tau, tau=2
        const float s = (h >= 1.0f) ? 1.0f : 0.0f; // spike(h - V_TH)
        spikes[tb * DD + d] = s;
        vs = h * (1.0f - s);                       // detached hard reset
    }
}

// ---------------------------------------------------------------------------
// Kernel 3: out[t,b,l,d] = q[t,b,l,d] * spikes[t,b,d]   (broadcast over L)
// float4 NT streams; spikes (64KB) stays hot in cache.
// N4 = 4,194,304 float4 elements -> 16384 blocks of 256.
// ---------------------------------------------------------------------------
__global__ __launch_bounds__(256) void snn_gate_kernel(
    const float* __restrict__ qp, const float* __restrict__ spikes,
    float* __restrict__ outp)
{
    const int i = blockIdx.x * blockDim.x + threadIdx.x;   // float4 index
    const int e  = i << 2;                                 // element index
    const int tb = e >> 18;                                // / (L*D), L*D=2^18
    const int d  = e & (DD - 1);                           // d base (mult of 4)

    const f4 s4 = *(const f4*)(spikes + tb * DD + d);      // cached broadcast
    const f4 q4 = __builtin_nontemporal_load((const f4*)(qp + e));
    __builtin_nontemporal_store(q4 * s4, (f4*)(outp + e));
}

// ---------------------------------------------------------------------------
extern "C" void kernel_launch(void* const* d_in, const int* in_sizes, int n_in,
                              void* d_out, int out_size, void* d_ws, size_t ws_size,
                              hipStream_t stream) {
    const float* q = (const float*)d_in[0];
    const float* k = (const float*)d_in[1];
    const float* v = (const float*)d_in[2];
    const unsigned char* mask = (const unsigned char*)d_in[3];  // bool tensor

    float* out  = (float*)d_out;                 // output: [T,B,L,D]
    float* attn = out + OUT_ELEMS;               // second tuple element: attn

    float* partial = (float*)d_ws;                       // [T*B, 8, D] = 512KB
    float* spikes  = partial + TT * BB * NCHUNK * DD;    // [T*B, D]    = 64KB

    // 8192 waves * 32 lanes / 256 threads = 1024 blocks
    snn_attn_partial_kernel<<<1024, 256, 0, stream>>>(k, v, mask, attn, partial);
    snn_lif_kernel<<<(BB * DD) / 256, 256, 0, stream>>>(partial, spikes);
    snn_gate_kernel<<<(OUT_ELEMS / 4) / 256, 256, 0, stream>>>(q, spikes, out);
}